// SMPLLayer_3040836845985
// MI455X (gfx1250) — compile-verified
//
#include <hip/hip_runtime.h>
#include <math.h>

#define NJ 24
#define NTHETA 72
#define NBETA 10
#define NVERT 6890
#define NBATCH 512
#define XSTRIDE 82
#define NCOLS (NVERT*3)      // 20670
#define KPOSE 207
#define PFSTRIDE 220         // padded A-operand: 207 pose | 0 | 10 betas | 0 0

// workspace layout (in floats)
#define S_OFF   0                            // 11*72 = 792
#define PF_OFF  800                          // 512*220
#define A_OFF   (800 + NBATCH*PFSTRIDE)      // 512*288

typedef float v2f __attribute__((ext_vector_type(2)));
typedef float v8f __attribute__((ext_vector_type(8)));
typedef unsigned int u32x4 __attribute__((ext_vector_type(4)));
typedef int i32x4 __attribute__((ext_vector_type(4)));
typedef int i32x8 __attribute__((ext_vector_type(8)));

#define WMMA_F32(a, b, c) \
  __builtin_amdgcn_wmma_f32_16x16x4_f32(false, (a), false, (b), (short)0, (c), false, false)

// ---------------------------------------------------------------------------
// TDM 2D tile load: global (row-major, stride0 elems/row) -> dense LDS tile.
// Descriptor bit-packing per CDNA5 ISA section 8.3-8.5 (D# groups).
// OOB columns (x >= td0) read as zero -> handles the ragged last vertex tile.
// ---------------------------------------------------------------------------
__device__ __forceinline__ void tdm_load_2d(unsigned lds_off, const void* gaddr,
                                            unsigned td0, unsigned td1,
                                            unsigned tile0, unsigned tile1,
                                            unsigned long long stride0) {
  unsigned long long ga = (unsigned long long)(uintptr_t)gaddr;
  u32x4 g0;
  g0[0] = 1u;                                             // count=1, user mode
  g0[1] = lds_off;                                        // lds_addr (bytes)
  g0[2] = (unsigned)ga;                                   // global_addr[31:0]
  g0[3] = (unsigned)((ga >> 32) & 0x01ffffffu) | (2u << 30); // addr[56:32] | type=2
  i32x8 g1;
  g1[0] = (int)(2u << 16);                                // data_size = 4 bytes
  g1[1] = (int)((td0 & 0xffffu) << 16);                   // tensor_dim0[15:0]
  g1[2] = (int)(((td0 >> 16) & 0xffffu) | ((td1 & 0xffffu) << 16));
  g1[3] = (int)(((td1 >> 16) & 0xffffu) | ((tile0 & 0xffffu) << 16));
  g1[4] = (int)(tile1 & 0xffffu);                         // tile_dim1 | tile_dim2=0
  g1[5] = (int)(stride0 & 0xffffffffull);                 // tensor_dim0_stride lo
  g1[6] = (int)((stride0 >> 32) & 0xffffull);             // stride hi | stride1=0
  g1[7] = 0;
  i32x4 gz = {0, 0, 0, 0};
#if __clang_major__ >= 23
  i32x8 gz8 = {0, 0, 0, 0, 0, 0, 0, 0};
  __builtin_amdgcn_tensor_load_to_lds(g0, g1, gz, gz, gz8, 0);
#else
  __builtin_amdgcn_tensor_load_to_lds(g0, g1, gz, gz, 0);
#endif
}

// ---------------------------------------------------------------------------
// K0: S[k,j,c] = sum_v Jreg[v,j] * (k<10 ? shapedirs[k, v*3+c] : v_template[v*3+c])
// One wave32 per output (11*72 = 792 outputs).
// ---------------------------------------------------------------------------
__global__ void smpl_reduce_S(const float* __restrict__ v_template,
                              const float* __restrict__ shapedirs,
                              const float* __restrict__ Jreg,
                              float* __restrict__ ws) {
  int wave = blockIdx.x * (blockDim.x >> 5) + (threadIdx.x >> 5);
  int lane = threadIdx.x & 31;
  if (wave >= 11 * 72) return;            // wave-uniform
  int k = wave / 72;
  int rem = wave % 72;
  int j = rem / 3;
  int c = rem % 3;
  float acc = 0.f;
  if (k < NBETA) {
    const float* src = shapedirs + (size_t)k * NCOLS;
    for (int v = lane; v < NVERT; v += 32)
      acc += Jreg[v * NJ + j] * src[v * 3 + c];
  } else {
    for (int v = lane; v < NVERT; v += 32)
      acc += Jreg[v * NJ + j] * v_template[v * 3 + c];
  }
#pragma unroll
  for (int off = 16; off > 0; off >>= 1)
    acc += __shfl_xor(acc, off, 32);
  if (lane == 0) ws[S_OFF + wave] = acc;
}

// ---------------------------------------------------------------------------
// K1: per-batch rodrigues + padded pose_feature row + joint regression + chain.
// ---------------------------------------------------------------------------
__global__ void smpl_pose_chain(const float* __restrict__ x,
                                const int* __restrict__ parents,
                                float* __restrict__ ws) {
  __shared__ float res[32][NJ * 12];      // per-thread global joint transforms
  int b = blockIdx.x * 32 + threadIdx.x;  // grid = 16 x 32 -> 512 exact
  const float* S = ws + S_OFF;
  float* pf   = ws + PF_OFF + (size_t)b * PFSTRIDE;
  float* Aout = ws + A_OFF + (size_t)b * (NJ * 12);
  const float* xb = x + (size_t)b * XSTRIDE;

  // J[b,j,c] = S[10,j,c] + sum_k betas[k]*S[k,j,c]
  float J[NJ * 3];
  for (int t = 0; t < 72; ++t) {
    float acc = S[NBETA * 72 + t];
    for (int k = 0; k < NBETA; ++k)
      acc += xb[NTHETA + k] * S[k * 72 + t];
    J[t] = acc;
  }

  // padded A-operand tail: [207]=0, [208..217]=betas, [218..219]=0
  pf[KPOSE] = 0.0f;
  for (int k = 0; k < NBETA; ++k) pf[208 + k] = xb[NTHETA + k];
  pf[218] = 0.0f;
  pf[219] = 0.0f;

  float* myres = res[threadIdx.x];
  for (int i = 0; i < NJ; ++i) {
    float tx = xb[i*3+0], ty = xb[i*3+1], tz = xb[i*3+2];
    float ax = tx + 1e-8f, ay = ty + 1e-8f, az = tz + 1e-8f;
    float ang = sqrtf(ax*ax + ay*ay + az*az);
    float inv = 1.0f / ang;
    float rx = tx*inv, ry = ty*inv, rz = tz*inv;
    float c = cosf(ang), s = sinf(ang), oc = 1.0f - c;
    float R[9];
    R[0] = c + oc*rx*rx;  R[1] = oc*rx*ry - s*rz; R[2] = oc*rx*rz + s*ry;
    R[3] = oc*rx*ry + s*rz; R[4] = c + oc*ry*ry; R[5] = oc*ry*rz - s*rx;
    R[6] = oc*rx*rz - s*ry; R[7] = oc*ry*rz + s*rx; R[8] = c + oc*rz*rz;

    if (i > 0) {
#pragma unroll
      for (int t = 0; t < 9; ++t)
        pf[(i-1)*9 + t] = R[t] - ((t==0 || t==4 || t==8) ? 1.0f : 0.0f);
    }

    if (i == 0) {
#pragma unroll
      for (int p = 0; p < 3; ++p) {
        myres[p*4+0] = R[p*3+0]; myres[p*4+1] = R[p*3+1];
        myres[p*4+2] = R[p*3+2]; myres[p*4+3] = J[p];
      }
    } else {
      int par = parents[i];
      const float* Rp = &myres[par * 12];
      float tr0 = J[i*3+0] - J[par*3+0];
      float tr1 = J[i*3+1] - J[par*3+1];
      float tr2 = J[i*3+2] - J[par*3+2];
      float* o = &myres[i * 12];
#pragma unroll
      for (int p = 0; p < 3; ++p) {
        float a0 = Rp[p*4+0], a1 = Rp[p*4+1], a2 = Rp[p*4+2];
        o[p*4+0] = a0*R[0] + a1*R[3] + a2*R[6];
        o[p*4+1] = a0*R[1] + a1*R[4] + a2*R[7];
        o[p*4+2] = a0*R[2] + a1*R[5] + a2*R[8];
        o[p*4+3] = a0*tr0 + a1*tr1 + a2*tr2 + Rp[p*4+3];
      }
    }
  }

  for (int jj = 0; jj < NJ; ++jj) {
    const float* G = &myres[jj * 12];
    float jx = J[jj*3+0], jy = J[jj*3+1], jz = J[jj*3+2];
#pragma unroll
    for (int p = 0; p < 3; ++p) {
      Aout[jj*12 + p*4 + 0] = G[p*4+0];
      Aout[jj*12 + p*4 + 1] = G[p*4+1];
      Aout[jj*12 + p*4 + 2] = G[p*4+2];
      Aout[jj*12 + p*4 + 3] = G[p*4+3] - (G[p*4+0]*jx + G[p*4+1]*jy + G[p*4+2]*jz);
    }
  }
}

// ---------------------------------------------------------------------------
// K2: TDM stages the 220x48 B panel into LDS; single GEMM wave runs a uniform
//     55-step WMMA loop (3 accumulators); then fused LBS skinning.
// ---------------------------------------------------------------------------
__global__ void smpl_verts(const float* __restrict__ v_template,
                           const float* __restrict__ shapedirs,
                           const float* __restrict__ posedirs,
                           const float* __restrict__ lbs_weights,
                           const float* __restrict__ ws,
                           float* __restrict__ out) {
  __shared__ float Bs[PFSTRIDE][48];  // B panel: posedirs rows 0..206, zero 207,
                                      //          shapedirs rows 208..217, zero 218-219
  __shared__ float vp[16][48];        // v_posed tile: [batch][16 verts * 3]
  __shared__ float wt[16][24];        // lbs weights for 16 verts
  __shared__ float At[16][NJ*12];     // joint transforms for 16 batches

  int vtile = blockIdx.x;   // 0..430
  int btile = blockIdx.y;   // 0..31
  int tid = threadIdx.x;    // 64 threads = 2 waves
  int lane = tid & 31;
  int wave = tid >> 5;
  int v0 = vtile * 16;
  int b0 = btile * 16;
  int cstart = v0 * 3;
  unsigned remc = (unsigned)(NCOLS - cstart);   // remaining columns (>= 30)

  if (wave == 1) {
    // zero the pad rows of the B panel (rows 207, 218, 219)
    for (int i = lane; i < 48; i += 32) {
      Bs[KPOSE][i] = 0.0f; Bs[218][i] = 0.0f; Bs[219][i] = 0.0f;
    }
    // stage lbs weights + A matrices (overlaps wave0's TDM + GEMM)
    for (int idx = lane; idx < 16 * 24; idx += 32) {
      int vi = idx / 24, j = idx % 24;
      int v = v0 + vi;
      wt[vi][j] = (v < NVERT) ? lbs_weights[v * NJ + j] : 0.0f;
    }
    const float* Asrc = ws + A_OFF;
    for (int idx = lane; idx < 16 * NJ * 12; idx += 32) {
      int bi = idx / (NJ * 12), t = idx % (NJ * 12);
      At[bi][t] = Asrc[(size_t)(b0 + bi) * (NJ * 12) + t];
    }
  } else {
    // ---- wave 0: TDM panel load + WMMA GEMM (EXEC all ones) ----
    unsigned bs_base = (unsigned)(uintptr_t)&Bs[0][0];
    tdm_load_2d(bs_base, posedirs + cstart, remc, KPOSE, 48, KPOSE,
                (unsigned long long)NCOLS);
    tdm_load_2d(bs_base + 208u * 48u * 4u, shapedirs + cstart, remc, NBETA,
                48, NBETA, (unsigned long long)NCOLS);

    int laneM = lane & 15;
    int kh = (lane < 16) ? 0 : 2;                 // half-wave K split
    const float* pfp = ws + PF_OFF + (size_t)(b0 + laneM) * PFSTRIDE + kh;

    __builtin_amdgcn_s_wait_tensorcnt(0);         // B panel resident in LDS

    v8f acc0 = {}, acc1 = {}, acc2 = {};
    for (int k = 0; k < PFSTRIDE; k += 4) {       // uniform 55-step K loop
      v2f a = *(const v2f*)(pfp + k);             // A frag: global_load_b64
      int r = k + kh;
      v2f vb;
      vb.x = Bs[r][laneM];      vb.y = Bs[r+1][laneM];      acc0 = WMMA_F32(a, vb, acc0);
      vb.x = Bs[r][laneM + 16]; vb.y = Bs[r+1][laneM + 16]; acc1 = WMMA_F32(a, vb, acc1);
      vb.x = Bs[r][laneM + 32]; vb.y = Bs[r+1][laneM + 32]; acc2 = WMMA_F32(a, vb, acc2);
    }

    // epilogue: + v_template (clamped cols for ragged edge), scatter to LDS
    int c0 = (cstart + laneM      < NCOLS) ? cstart + laneM      : (NCOLS - 1);
    int c1 = (cstart + laneM + 16 < NCOLS) ? cstart + laneM + 16 : (NCOLS - 1);
    int c2 = (cstart + laneM + 32 < NCOLS) ? cstart + laneM + 32 : (NCOLS - 1);
    float vt0 = v_template[c0], vt1 = v_template[c1], vt2 = v_template[c2];
    int mh = (lane < 16) ? 0 : 8;
#pragma unroll
    for (int r = 0; r < 8; ++r) {
      int m = r + mh;
      vp[m][laneM]      = acc0[r] + vt0;
      vp[m][laneM + 16] = acc1[r] + vt1;
      vp[m][laneM + 32] = acc2[r] + vt2;
    }
  }
  __syncthreads();

  // LBS skinning: 256 (batch,vert) pairs, 4 per thread
  for (int pair = tid; pair < 256; pair += 64) {
    int bi = pair >> 4, vi = pair & 15;
    int v = v0 + vi;
    if (v >= NVERT) continue;
    float px = vp[bi][vi*3+0], py = vp[bi][vi*3+1], pz = vp[bi][vi*3+2];
    float o0 = 0.f, o1 = 0.f, o2 = 0.f;
    const float* Ab = At[bi];
#pragma unroll
    for (int j = 0; j < NJ; ++j) {
      float wj = wt[vi][j];
      const float* G = &Ab[j * 12];
      o0 += wj * (G[0]*px + G[1]*py + G[2]*pz  + G[3]);
      o1 += wj * (G[4]*px + G[5]*py + G[6]*pz  + G[7]);
      o2 += wj * (G[8]*px + G[9]*py + G[10]*pz + G[11]);
    }
    size_t o = ((size_t)(b0 + bi) * NVERT + v) * 3;
    out[o+0] = o0; out[o+1] = o1; out[o+2] = o2;
  }
}

// ---------------------------------------------------------------------------
extern "C" void kernel_launch(void* const* d_in, const int* in_sizes, int n_in,
                              void* d_out, int out_size, void* d_ws, size_t ws_size,
                              hipStream_t stream) {
  const float* xin        = (const float*)d_in[0];
  const float* v_template = (const float*)d_in[1];
  const float* shapedirs  = (const float*)d_in[2];
  const float* J_reg      = (const float*)d_in[3];
  const float* posedirs   = (const float*)d_in[4];
  const float* lbs        = (const float*)d_in[5];
  const int*   parents    = (const int*)d_in[6];
  float* out = (float*)d_out;
  float* ws  = (float*)d_ws;

  // K0: 792 wave-reductions (99 blocks x 8 waves)
  smpl_reduce_S<<<99, 256, 0, stream>>>(v_template, shapedirs, J_reg, ws);
  // K1: one thread per batch (512 total)
  smpl_pose_chain<<<16, 32, 0, stream>>>(xin, parents, ws);
  // K2: TDM-staged WMMA GEMM + skinning; 431 vert-tiles x 32 batch-tiles
  smpl_verts<<<dim3(431, 32), 64, 0, stream>>>(v_template, shapedirs, posedirs,
                                               lbs, ws, out);
}